// MCRGANloss_2276332667592
// MI455X (gfx1250) — compile-verified
//
#include <hip/hip_runtime.h>
#include <stddef.h>
#include <stdint.h>

typedef float v2f __attribute__((ext_vector_type(2)));
typedef float v8f __attribute__((ext_vector_type(8)));

#define CLS 10
#define EPSC 0.5f
#define KB 32          // K-chunk staged in LDS (gram kernel)
#define BT 64          // output tile edge per workgroup (gram kernel)
#define LDP 72         // LDS row stride (floats); 288B rows stay 16B aligned,
                       // 2*72 mod 64 = 16 -> half-wave fragment reads on disjoint banks
#define NB 32          // LDL^T panel width
#define MAXD 512       // max matrix dim supported by the blocked LDL panels
#define PLP 33         // panel LDS row stride (floats)

// ---------------- class counts + exclusive offsets ----------------
__global__ void count_kernel(const int* __restrict__ lab, int n,
                             int* __restrict__ counts, int* __restrict__ offs) {
    __shared__ int bins[CLS];
    const int tid = threadIdx.x;
    if (tid < CLS) bins[tid] = 0;
    __syncthreads();
    for (int i = tid; i < n; i += blockDim.x) {
        const int c = lab[i];
        if (c >= 0 && c < CLS) atomicAdd(&bins[c], 1);
    }
    __syncthreads();
    if (tid == 0) {
        int run = 0;
        for (int c = 0; c < CLS; ++c) { counts[c] = bins[c]; offs[c] = run; run += bins[c]; }
        offs[CLS] = run;
    }
}

// ---------------- deterministic class-sorted permutation ----------------
__global__ void build_perm_kernel(const int* __restrict__ lab, const int* __restrict__ offs,
                                  int n, int* __restrict__ perm) {
    const int c = threadIdx.x;
    if (c >= CLS) return;
    int cur = offs[c];
    for (int i = 0; i < n; ++i)
        if (lab[i] == c) perm[cur++] = i;
}

// ---------------- per-class grams: f32 WMMA + double-buffered async->LDS ----------------
struct GramSM {
    float As[2][KB][LDP];
    float Bs[2][KB][LDP];
};

__global__ void gram_kernel(const float* __restrict__ Z, const float* __restrict__ Zb,
                            const int* __restrict__ perm, const int* __restrict__ offs,
                            float* __restrict__ Gz, float* __restrict__ Gzb,
                            int n, int d) {
    __shared__ __align__(16) GramSM sm;

    const int job = blockIdx.y;
    const int c   = job % CLS;
    const float* __restrict__ src = (job < CLS) ? Z : Zb;
    float* __restrict__ out = ((job < CLS) ? Gz : Gzb) + (size_t)c * d * d;

    const int tpr  = d / BT;
    const int rowb = (blockIdx.x / tpr) * BT;
    const int colb = (blockIdx.x % tpr) * BT;

    const int tid  = threadIdx.x;       // 0..127
    const int wave = tid >> 5;          // 0..3 -> 16-row M strip within tile
    const int lane = tid & 31;
    const int lo   = lane & 15;         // M for A frag, N for B frag
    const int hi   = lane >> 4;         // K pair select {0,1} vs {2,3}

    const int off = offs[c];
    const int end = offs[c + 1];

    const unsigned asBase = (unsigned)offsetof(GramSM, As);
    const unsigned bsBase = (unsigned)offsetof(GramSM, Bs);

    v8f acc0 = {}, acc1 = {}, acc2 = {}, acc3 = {};

    if (off < end) {
        const int kmax = end - 1;

        // one chunk fill = 8 async b128 per thread (4 A-quads + 4 B-quads);
        // 8 ASYNCcnt increments per wave, completion in-order.
        auto prefetch = [&](int buf, int k0) {
            #pragma unroll
            for (int j = 0; j < 4; ++j) {
                const int q  = tid + 128 * j;
                const int r  = q >> 4;
                const int c4 = (q & 15) << 2;
                int k = k0 + r;
                k = (k < off) ? off : ((k > kmax) ? kmax : k);   // clamp: finite data
                const size_t srow = (size_t)perm[k] * (size_t)d;
                const unsigned lofs = (unsigned)(((buf * KB + r) * LDP + c4) * 4);
                const uint64_t ga = (uint64_t)(uintptr_t)(src + srow + rowb + c4);
                const uint64_t gb = (uint64_t)(uintptr_t)(src + srow + colb + c4);
                asm volatile("global_load_async_to_lds_b128 %0, %1, off"
                             :: "v"(asBase + lofs), "v"(ga) : "memory");
                asm volatile("global_load_async_to_lds_b128 %0, %1, off"
                             :: "v"(bsBase + lofs), "v"(gb) : "memory");
            }
        };

        const int k0start = off & ~(KB - 1);
        prefetch(0, k0start);

        int bufI = 0;
        for (int k0 = k0start; k0 < end; k0 += KB, bufI ^= 1) {
            const int buf = bufI;
            if ((k0 + KB) < end) {
                prefetch(buf ^ 1, k0 + KB);
                asm volatile("s_wait_asynccnt 0x8" ::: "memory");   // oldest 8 = this buf
            } else {
                asm volatile("s_wait_asynccnt 0x0" ::: "memory");
            }
            __syncthreads();

            if ((k0 < off) | (k0 + KB > end)) {      // boundary chunk: zero invalid A rows
                for (int idx = tid; idx < KB * BT; idx += 128) {
                    const int r = idx / BT;
                    const int k = k0 + r;
                    if (k < off || k >= end) sm.As[buf][r][idx % BT] = 0.0f;
                }
                __syncthreads();
            }

            const float (*Asb)[LDP] = sm.As[buf];
            const float (*Bsb)[LDP] = sm.Bs[buf];
            #pragma unroll
            for (int ks = 0; ks < KB; ks += 4) {
                const int kA = ks + 2 * hi;
                v2f a, b0, b1, b2, b3;
                a.x  = Asb[kA][wave * 16 + lo];
                a.y  = Asb[kA + 1][wave * 16 + lo];
                b0.x = Bsb[kA][ 0 + lo];  b0.y = Bsb[kA + 1][ 0 + lo];
                b1.x = Bsb[kA][16 + lo];  b1.y = Bsb[kA + 1][16 + lo];
                b2.x = Bsb[kA][32 + lo];  b2.y = Bsb[kA + 1][32 + lo];
                b3.x = Bsb[kA][48 + lo];  b3.y = Bsb[kA + 1][48 + lo];
                acc0 = __builtin_amdgcn_wmma_f32_16x16x4_f32(false, a, false, b0, (short)0, acc0, false, false);
                acc1 = __builtin_amdgcn_wmma_f32_16x16x4_f32(false, a, false, b1, (short)0, acc1, false, false);
                acc2 = __builtin_amdgcn_wmma_f32_16x16x4_f32(false, a, false, b2, (short)0, acc2, false, false);
                acc3 = __builtin_amdgcn_wmma_f32_16x16x4_f32(false, a, false, b3, (short)0, acc3, false, false);
            }
            __syncthreads();
        }
    }

    #pragma unroll
    for (int v = 0; v < 8; ++v) {
        const size_t row = (size_t)(rowb + wave * 16 + v + 8 * hi) * d;
        out[row + colb +  0 + lo] = acc0[v];
        out[row + colb + 16 + lo] = acc1[v];
        out[row + colb + 32 + lo] = acc2[v];
        out[row + colb + 48 + lo] = acc3[v];
    }
}

// ---------------- build the 32 SPD matrices  M = I + scale * Gram ----------------
__global__ void assemble_kernel(float* __restrict__ Gz, float* __restrict__ Gzb,
                                float* __restrict__ Cross, float* __restrict__ Dz,
                                float* __restrict__ Dzb, const int* __restrict__ counts,
                                int n, int d) {
    const size_t mat = (size_t)d * d;
    const size_t idx = (size_t)blockIdx.x * blockDim.x + threadIdx.x;
    if (idx >= mat) return;
    const int i = (int)(idx / d);
    const int j = (int)(idx % d);
    const float eye   = (i == j) ? 1.0f : 0.0f;
    const float sdisc = (float)d / ((float)n * EPSC);
    float sz = 0.0f, szb = 0.0f;
    #pragma unroll
    for (int c = 0; c < CLS; ++c) {
        const float gz  = Gz [c * mat + idx];
        const float gzb = Gzb[c * mat + idx];
        sz += gz; szb += gzb;
        const float cnt   = (float)counts[c];
        const float scalz = (float)d / ((cnt + 1e-8f) * EPSC);
        const float scalx = (cnt > 0.0f) ? (float)d / (2.0f * cnt * EPSC) : 0.0f;
        Cross[c * mat + idx] = eye + scalx * (gz + gzb);
        Gz  [c * mat + idx]  = eye + scalz * gz;
        Gzb [c * mat + idx]  = eye + scalz * gzb;
    }
    Dz [idx] = eye + sdisc * sz;
    Dzb[idx] = eye + sdisc * szb;
}

// ---------------- blocked LDL^T logdet with WMMA trailing update ----------------
// One workgroup (256 threads = 8 waves) per matrix. NB=32 panels.
// Only the lower triangle is maintained; the diagonal block is mirrored on load.
// Unscaled column values Traw[p][q] = d_q * L[p][q], so no unit-L writeback is needed.
struct LdlSM {
    float Tb[NB][PLP];                 // diagonal block (raw LDL workspace)
    float Lp[MAXD - NB][PLP];          // panel L rows (below the diagonal block)
    float Wp[MAXD - NB][PLP];          // panel W = -L*D rows
};

__global__ void ldl_logdet_kernel(float* __restrict__ Gz, float* __restrict__ Gzb,
                                  float* __restrict__ Cross, float* __restrict__ Dz,
                                  float* __restrict__ Dzb, float* __restrict__ ldout,
                                  int d) {
    __shared__ LdlSM sm;
    const int m = blockIdx.x;
    const size_t mat = (size_t)d * d;
    float* A;
    if      (m == 0)           A = Dz;
    else if (m == 1)           A = Dzb;
    else if (m < 2 + CLS)      A = Gz    + (size_t)(m - 2)           * mat;
    else if (m < 2 + 2 * CLS)  A = Gzb   + (size_t)(m - 2 - CLS)     * mat;
    else                       A = Cross + (size_t)(m - 2 - 2 * CLS) * mat;

    const int tid  = threadIdx.x;       // 0..255
    const int wv   = tid >> 5;          // wave 0..7
    const int lane = tid & 31;
    const int lo   = lane & 15;
    const int hi   = lane >> 4;

    float logacc = 0.0f;

    for (int kb = 0; kb < d; kb += NB) {
        const int kbp = kb + NB;        // first row below the diagonal block

        // ---- 1) load diagonal block (mirror lower triangle) and factor (raw LDL) ----
        for (int idx = tid; idx < NB * NB; idx += 256) {
            const int bi = idx / NB, bj = idx % NB;
            const int si = bi >= bj ? bi : bj;
            const int sj = bi >= bj ? bj : bi;
            sm.Tb[bi][bj] = A[(size_t)(kb + si) * d + kb + sj];
        }
        __syncthreads();
        #pragma unroll 1
        for (int p = 0; p < NB; ++p) {
            if (tid == 0) logacc += logf(sm.Tb[p][p]);
            if (tid > p && tid < NB) {
                const float lip = sm.Tb[tid][p] / sm.Tb[p][p];
                for (int j = p + 1; j <= tid; ++j)
                    sm.Tb[tid][j] -= lip * sm.Tb[j][p];   // column p is never written here
            }
            __syncthreads();
        }

        const int below = d - kbp;
        if (below > 0) {
            // ---- 2) panel solve: rows i >= kbp; lrow[p] = (a_p - sum_{q<p} lrow_q*Traw[p][q]) / d_p
            for (int i = kbp + tid; i < d; i += 256) {
                const int rel = i - kbp;
                const float* __restrict__ arow = A + (size_t)i * d + kb;
                float lrow[NB];
                #pragma unroll
                for (int p = 0; p < NB; ++p) {
                    float s = arow[p];
                    #pragma unroll
                    for (int q = 0; q < p; ++q) s -= lrow[q] * sm.Tb[p][q];
                    lrow[p] = s / sm.Tb[p][p];
                }
                #pragma unroll
                for (int p = 0; p < NB; ++p) {
                    sm.Lp[rel][p] = lrow[p];
                    sm.Wp[rel][p] = -lrow[p] * sm.Tb[p][p];
                }
            }
            __syncthreads();

            // ---- 3) trailing update A += L * W^T on lower-tri 16x16 tiles via WMMA ----
            const int Ts = below >> 4;
            int idx = 0;
            for (int ti = 0; ti < Ts; ++ti) {
                for (int tj = 0; tj <= ti; ++tj, ++idx) {
                    if ((idx & 7) != wv) continue;      // wave-uniform distribution
                    const int i0 = kbp + ti * 16;
                    const int j0 = kbp + tj * 16;
                    const int ra = ti * 16 + lo;        // A-frag M row (rel)
                    const int rb = tj * 16 + lo;        // B-frag N row (rel)
                    v8f acc;
                    #pragma unroll
                    for (int v = 0; v < 8; ++v)
                        acc[v] = A[(size_t)(i0 + v + 8 * hi) * d + j0 + lo];
                    #pragma unroll
                    for (int ks = 0; ks < NB; ks += 4) {
                        const int kk = ks + 2 * hi;
                        v2f a, b;
                        a.x = sm.Lp[ra][kk];  a.y = sm.Lp[ra][kk + 1];
                        b.x = sm.Wp[rb][kk];  b.y = sm.Wp[rb][kk + 1];
                        acc = __builtin_amdgcn_wmma_f32_16x16x4_f32(
                            false, a, false, b, (short)0, acc, false, false);
                    }
                    #pragma unroll
                    for (int v = 0; v < 8; ++v)
                        A[(size_t)(i0 + v + 8 * hi) * d + j0 + lo] = acc[v];
                }
            }
            __syncthreads();
        }
    }
    if (tid == 0) ldout[m] = logacc;
}

// ---------------- final scalar combination ----------------
__global__ void final_kernel(const float* __restrict__ ld, const int* __restrict__ counts,
                             float* __restrict__ outp, int n) {
    if (threadIdx.x != 0 || blockIdx.x != 0) return;
    const float z_disc  = ld[0] * 0.5f;
    const float zb_disc = ld[1] * 0.5f;
    float z_comp = 0.f, zb_comp = 0.f, sum_z = 0.f, sum_zb = 0.f, itemR = 0.f;
    for (int c = 0; c < CLS; ++c) {
        const float cnt = (float)counts[c];
        const float zl  = (cnt == 0.0f) ? 1.0f : ld[2 + c];
        const float zbl = (cnt == 0.0f) ? 1.0f : ld[2 + CLS + c];
        const float sc  = (cnt + 1e-8f) / (2.0f * (float)n);
        z_comp += sc * zl;
        zb_comp += sc * zbl;
        sum_z  += zl;
        sum_zb += zbl;
        itemR  += ld[2 + 2 * CLS + c] * 0.5f;
    }
    const float t1 = z_disc  - z_comp;
    const float t2 = zb_disc - zb_comp;
    const float t3 = itemR - 0.25f * sum_z - 0.25f * sum_zb;
    outp[0] = -(t1 + t2 + t3);                          // errD
    outp[1] = -t1;                                      // z_total
    outp[2] = -t2;                                      // zbar_total
    outp[3] = -itemR + 0.25f * sum_z + 0.25f * sum_zb;  // term3
}

extern "C" void kernel_launch(void* const* d_in, const int* in_sizes, int n_in,
                              void* d_out, int out_size, void* d_ws, size_t ws_size,
                              hipStream_t stream) {
    const float* Z   = (const float*)d_in[0];
    const float* Zb  = (const float*)d_in[1];
    const int*   lab = (const int*)d_in[2];
    const int n = in_sizes[2];          // 8192
    const int d = in_sizes[0] / n;      // 512 (blocked LDL assumes d%32==0, d<=MAXD)
    const size_t mat = (size_t)d * d;

    // ws layout (floats): Gz[10*mat] Gzb[10*mat] Cross[10*mat] Dz[mat] Dzb[mat]
    //                     ld[64] | ints: counts[16] offs[16] perm[n]
    float* ws    = (float*)d_ws;
    float* Gz    = ws;
    float* Gzb   = Gz    + (size_t)CLS * mat;
    float* Cross = Gzb   + (size_t)CLS * mat;
    float* Dz    = Cross + (size_t)CLS * mat;
    float* Dzb   = Dz    + mat;
    float* ldts  = Dzb   + mat;
    int*   cnts  = (int*)(ldts + 64);
    int*   offs  = cnts + 16;
    int*   perm  = offs + 16;

    count_kernel<<<1, 1024, 0, stream>>>(lab, n, cnts, offs);
    build_perm_kernel<<<1, 32, 0, stream>>>(lab, offs, n, perm);

    const int tiles = (d / BT) * (d / BT);
    dim3 grid(tiles, 2 * CLS);
    gram_kernel<<<grid, 128, 0, stream>>>(Z, Zb, perm, offs, Gz, Gzb, n, d);

    const int tot = (int)((mat + 255) / 256);
    assemble_kernel<<<tot, 256, 0, stream>>>(Gz, Gzb, Cross, Dz, Dzb, cnts, n, d);

    ldl_logdet_kernel<<<2 + 3 * CLS, 256, 0, stream>>>(Gz, Gzb, Cross, Dz, Dzb, ldts, d);

    final_kernel<<<1, 32, 0, stream>>>(ldts, cnts, (float*)d_out, n);
}